// TSelfRegLoss_88656714924224
// MI455X (gfx1250) — compile-verified
//
#include <hip/hip_runtime.h>
#include <math.h>

#define BATCH    4096
#define KDIM     32000
#define KVEC     (KDIM / 4)           // 8000 float4 per row
#define NTHREADS 256
#define NCHUNK   4
#define CHUNK_ELEMS (KDIM / NCHUNK)   // 8000 floats = 32000 bytes per TDM descriptor
#define CHUNK_VECS  (CHUNK_ELEMS / 4) // 2000 float4 per chunk

typedef unsigned int v4u __attribute__((ext_vector_type(4)));
typedef int          v8i __attribute__((ext_vector_type(8)));
typedef int          v4i __attribute__((ext_vector_type(4)));

#if defined(__gfx1250__) && __has_builtin(__builtin_amdgcn_tensor_load_to_lds)
#define HAVE_TDM 1
#else
#define HAVE_TDM 0
#endif

__device__ __forceinline__ float wave_max(float v) {
#pragma unroll
  for (int o = 16; o > 0; o >>= 1) v = fmaxf(v, __shfl_xor(v, o, 32));
  return v;
}
__device__ __forceinline__ float wave_sum(float v) {
#pragma unroll
  for (int o = 16; o > 0; o >>= 1) v += __shfl_xor(v, o, 32);
  return v;
}

template <int N>
__device__ __forceinline__ void wait_tensor() {
#if HAVE_TDM
  __builtin_amdgcn_s_wait_tensorcnt(N);
#endif
}

// One block per row: stage the 128KB row into LDS with 4 chunked TDM
// descriptors (compute on chunk c overlaps DMA of chunks c+1..), then two
// softmax passes from LDS via ds_load_b128. Emit per-row lp_label and
// S = sum_k log_probs.
__global__ __launch_bounds__(NTHREADS)
void TSelfRegLoss_rowstats(const float* __restrict__ logits,
                           const int*  __restrict__ labels,
                           float* __restrict__ lp_out,
                           float* __restrict__ S_out) {
  const int row = blockIdx.x;
  const int tid = threadIdx.x;

  __shared__ __align__(16) float rowbuf[KDIM];          // 128000 B (fits 320KB/WGP LDS)
  __shared__ float s_m[NTHREADS / 32], s_t[NTHREADS / 32], s_e[NTHREADS / 32];

#if HAVE_TDM
  if (tid < 32) {  // wave 0 issues 4 chunked tensor-DMA descriptors (EXEC ignored)
    const unsigned long long gbase =
        (unsigned long long)(uintptr_t)(logits + (size_t)row * KDIM);
    // Generic LDS pointer: low 32 bits are the workgroup-relative LDS byte offset.
    const unsigned int lds0 = (unsigned int)(uintptr_t)(void*)rowbuf;
#pragma unroll
    for (int c = 0; c < NCHUNK; ++c) {
      const unsigned long long ga = gbase + (unsigned long long)c * (CHUNK_ELEMS * 4ull);
      const unsigned int       la = lds0 + (unsigned int)(c * CHUNK_ELEMS * 4);
      v4u g0;
      g0[0] = 1u;                                           // count=1 (valid), user mode
      g0[1] = la;                                           // lds_addr  @ bits [63:32]
      g0[2] = (unsigned int)(ga & 0xFFFFFFFFull);           // global_addr[31:0]
      g0[3] = (unsigned int)((ga >> 32) & 0x1FFFFFFull)     // global_addr[56:32]
              | 0x80000000u;                                // type=2 ("image") @ [127:126]
      v8i g1;
      g1[0] = 0x00020000;                                   // wg_mask=0, data_size=2 (4B)
      g1[1] = (CHUNK_ELEMS & 0xFFFF) << 16;                 // tensor_dim0[15:0] @ [63:48]
      g1[2] = ((CHUNK_ELEMS >> 16) & 0xFFFF) | (1 << 16);   // dim0[31:16] | tensor_dim1=1
      g1[3] = (CHUNK_ELEMS & 0xFFFF) << 16;                 // tile_dim0 @ [127:112]
      g1[4] = 1;                                            // tile_dim1=1, tile_dim2=0
      g1[5] = CHUNK_ELEMS;                                  // tensor_dim0_stride[31:0]
      g1[6] = (CHUNK_ELEMS & 0xFFFF) << 16;                 // tensor_dim1_stride[15:0]
      g1[7] = 0;
      v4i z4 = {0, 0, 0, 0};
      v8i z8 = {0, 0, 0, 0, 0, 0, 0, 0};
      __builtin_amdgcn_tensor_load_to_lds(g0, g1, z4, z4, z8, 0);
    }
  }
#else
  {  // fallback staging path
    const float* __restrict__ g = logits + (size_t)row * KDIM;
    for (int i = tid; i < KDIM; i += NTHREADS) rowbuf[i] = g[i];
  }
#endif

  const float4* __restrict__ rb4 = reinterpret_cast<const float4*>(rowbuf);

  // Pass 1: row max + sum of logits, chunk-pipelined behind the TDM.
  // 4 independent accumulator lanes shorten the dependency chain and map to
  // packed/dual-issue f32 ops.
  float4 m4 = make_float4(-3.402823466e38f, -3.402823466e38f,
                          -3.402823466e38f, -3.402823466e38f);
  float4 t4 = make_float4(0.f, 0.f, 0.f, 0.f);
  auto consume = [&](int c) {
    for (int i = c * CHUNK_VECS + tid; i < (c + 1) * CHUNK_VECS; i += NTHREADS) {
      const float4 v = rb4[i];                 // ds_load_b128
      m4.x = fmaxf(m4.x, v.x); t4.x += v.x;
      m4.y = fmaxf(m4.y, v.y); t4.y += v.y;
      m4.z = fmaxf(m4.z, v.z); t4.z += v.z;
      m4.w = fmaxf(m4.w, v.w); t4.w += v.w;
    }
  };
  if (tid < 32) wait_tensor<NCHUNK - 1>(); __syncthreads(); consume(0);
  if (tid < 32) wait_tensor<NCHUNK - 2>(); __syncthreads(); consume(1);
  if (tid < 32) wait_tensor<NCHUNK - 3>(); __syncthreads(); consume(2);
  if (tid < 32) wait_tensor<0>();          __syncthreads(); consume(3);

  const float m    = fmaxf(fmaxf(m4.x, m4.y), fmaxf(m4.z, m4.w));
  const float tsum = (t4.x + t4.y) + (t4.z + t4.w);

  const int wid = tid >> 5, lane = tid & 31;
  const float wm = wave_max(m);
  const float wt = wave_sum(tsum);
  if (lane == 0) { s_m[wid] = wm; s_t[wid] = wt; }
  __syncthreads();
  float bm = s_m[0], bt = 0.0f;
#pragma unroll
  for (int w = 0; w < NTHREADS / 32; ++w) { bm = fmaxf(bm, s_m[w]); bt += s_t[w]; }

  // Pass 2: sum of exp(x - max) from LDS (v_exp_f32 via exp2), vectorized.
  const float log2e   = 1.4426950408889634f;
  const float bm_l2e  = bm * log2e;
  float4 e4 = make_float4(0.f, 0.f, 0.f, 0.f);
  for (int i = tid; i < KVEC; i += NTHREADS) {
    const float4 v = rb4[i];                   // ds_load_b128
    e4.x += exp2f(fmaf(v.x, log2e, -bm_l2e));
    e4.y += exp2f(fmaf(v.y, log2e, -bm_l2e));
    e4.z += exp2f(fmaf(v.z, log2e, -bm_l2e));
    e4.w += exp2f(fmaf(v.w, log2e, -bm_l2e));
  }
  const float we = wave_sum((e4.x + e4.y) + (e4.z + e4.w));
  if (lane == 0) s_e[wid] = we;
  __syncthreads();

  if (tid == 0) {
    float es = 0.0f;
#pragma unroll
    for (int w = 0; w < NTHREADS / 32; ++w) es += s_e[w];
    const float lse = bm + logf(es);
    const int   lab = labels[row];
    lp_out[row] = rowbuf[lab] - lse;               // log_prob at label
    S_out[row]  = bt - (float)KDIM * lse;          // sum_k log_probs
  }
}

__global__ __launch_bounds__(NTHREADS)
void TSelfRegLoss_finalize(const float* __restrict__ lp,
                           const float* __restrict__ S,
                           float* __restrict__ out,
                           float p_hot, float p_off, float C0) {
  const int tid = threadIdx.x;
  float sl = 0.0f, ss = 0.0f;
  for (int i = tid; i < BATCH; i += NTHREADS) { sl += lp[i]; ss += S[i]; }
  sl = wave_sum(sl);
  ss = wave_sum(ss);
  __shared__ float shl[NTHREADS / 32], shs[NTHREADS / 32];
  const int wid = tid >> 5, lane = tid & 31;
  if (lane == 0) { shl[wid] = sl; shs[wid] = ss; }
  __syncthreads();
  if (tid == 0) {
    float SL = 0.0f, SS = 0.0f;
#pragma unroll
    for (int w = 0; w < NTHREADS / 32; ++w) { SL += shl[w]; SS += shs[w]; }
    const float kl_sum  = (float)BATCH * C0 - (p_hot - p_off) * SL - p_off * SS;
    const float loss_ce = -SL / (float)BATCH;
    const float kl      = kl_sum / ((float)BATCH * (float)KDIM);
    out[0] = 0.05f * loss_ce + 0.95f * kl;   // (1-ALPHA)*ce + ALPHA*kl*MULTIPLIER
  }
}

extern "C" void kernel_launch(void* const* d_in, const int* in_sizes, int n_in,
                              void* d_out, int out_size, void* d_ws, size_t ws_size,
                              hipStream_t stream) {
  (void)in_sizes; (void)n_in; (void)out_size; (void)ws_size;
  const float* logits = (const float*)d_in[0];
  const int*   labels = (const int*)d_in[1];
  float* lp = (float*)d_ws;
  float* Ss = lp + BATCH;

  // Teacher-distribution constants (data-independent) in double on host.
  const double Kd      = (double)KDIM;
  const double off_val = (1.0 - 0.99) / (Kd - 1.0);
  const double t_hot   = 0.99 / 20.0;
  const double t_off   = off_val / 20.0;
  const double e_hot   = exp(t_hot), e_off = exp(t_off);
  const double denom   = e_hot + (Kd - 1.0) * e_off;
  const double p_hot   = e_hot / denom;
  const double p_off   = e_off / denom;
  const double C0      = p_hot * log(p_hot) + (Kd - 1.0) * p_off * log(p_off);

  TSelfRegLoss_rowstats<<<BATCH, NTHREADS, 0, stream>>>(logits, labels, lp, Ss);
  TSelfRegLoss_finalize<<<1, NTHREADS, 0, stream>>>(lp, Ss, (float*)d_out,
                                                    (float)p_hot, (float)p_off, (float)C0);
}